// DurationEncoding_2714419331616
// MI455X (gfx1250) — compile-verified
//
#include <hip/hip_runtime.h>

// DurationEncoding: searchsorted(edges[101], v) -> gather embed_table[101,128]
// Memory-bound: 512MB NT stores dominate (~22us floor @ 23.3 TB/s).
// CDNA5 path: TDM tensor_load_to_lds stages the 51.7KB table in LDS once per
// block; gathers come from LDS (ds_load_b128), stores are 128-bit NT.
// Bucket broadcast uses v_readlane (uniform lane index) to keep the LDS pipe
// free for the table reads.

#define NUM_BINS 101
#define DIM 128

typedef unsigned int u32x4 __attribute__((ext_vector_type(4)));
typedef int          i32x4 __attribute__((ext_vector_type(4)));
typedef int          i32x8 __attribute__((ext_vector_type(8)));
typedef float        f32x4 __attribute__((ext_vector_type(4)));

__global__ __launch_bounds__(256) void duration_embed_kernel(
    const float* __restrict__ time_value,
    const float* __restrict__ bin_edges,
    const float* __restrict__ embed_table,
    float* __restrict__ out,
    int n)
{
    __shared__ __align__(16) float lds_table[NUM_BINS * DIM];   // 51712 B
    __shared__ float lds_edges[NUM_BINS];

    const int tid = threadIdx.x;

    // Edges -> LDS (tiny, plain cooperative load).
    for (int i = tid; i < NUM_BINS; i += 256) lds_edges[i] = bin_edges[i];

    // Table -> LDS via Tensor Data Mover (one wave issues the DMA).
    if (tid == 0) {
        unsigned long long ga = (unsigned long long)(uintptr_t)embed_table;
        // Low 32 bits of a generic pointer to LDS == LDS byte address.
        unsigned int lds_off = (unsigned int)(uintptr_t)&lds_table[0];

        u32x4 g0;
        g0[0] = 1u;                                   // count=1, user descriptor
        g0[1] = lds_off;                              // lds_addr [63:32]
        g0[2] = (unsigned int)ga;                     // global_addr[31:0]
        g0[3] = ((unsigned int)(ga >> 32) & 0x01FFFFFFu) | (2u << 30); // addr[56:32], type=2

        i32x8 g1;
        g1[0] = 0x00020000;                           // workgroup_mask=0, data_size=2 (4B)
        g1[1] = (DIM & 0xFFFF) << 16;                 // tensor_dim0 = 128 (bits 79:48, low half)
        g1[2] = (NUM_BINS & 0xFFFF) << 16;            // tensor_dim0 hi=0; tensor_dim1 = 101 (low half)
        g1[3] = (DIM & 0xFFFF) << 16;                 // tensor_dim1 hi=0; tile_dim0 = 128
        g1[4] = NUM_BINS;                             // tile_dim1 = 101; tile_dim2 = 0
        g1[5] = DIM;                                  // tensor_dim0_stride = 128 (low 32)
        g1[6] = 0;                                    // stride0 hi; tensor_dim1_stride lo
        g1[7] = 0;                                    // tensor_dim1_stride hi

        i32x4 gz4 = {0, 0, 0, 0};                     // groups 2/3 unused (2D tile)
        i32x8 gz8 = {0, 0, 0, 0, 0, 0, 0, 0};        // extra group (unused)

        __builtin_amdgcn_tensor_load_to_lds(g0, g1, gz4, gz4, gz8, 0);
        __builtin_amdgcn_s_wait_tensorcnt(0);
    }
    __syncthreads();

    const int lane = tid & 31;
    const long long wid = (long long)blockIdx.x * 8 + (tid >> 5);
    const long long nw  = (long long)gridDim.x * 8;

    for (long long base = wid * 32; base < n; base += nw * 32) {
        // Lane-parallel bucketing: lane L handles row base+L.
        long long myrow = base + lane;
        float v = 0.0f;
        if (myrow < n) v = time_value[myrow];         // coalesced 128B/wave

        // Branchless lower_bound over 101 edges: count of edges < v.
        int pos = 0;
        #pragma unroll
        for (int step = 64; step >= 1; step >>= 1) {
            int np = pos + step;
            if (np <= NUM_BINS && lds_edges[np - 1] < v) pos = np;
        }
        int bucket = pos > (NUM_BINS - 1) ? (NUM_BINS - 1) : pos;  // jnp.take clips OOB

        long long rem = n - base;
        int limit = rem < 32 ? (int)rem : 32;
        for (int r = 0; r < limit; ++r) {
            // Uniform lane index -> v_readlane_b32 (SGPR bucket), not ds_bpermute.
            int b = __builtin_amdgcn_readlane(bucket, r);
            // Wave copies one 512B row: ds_load_b128 + NT global_store_b128.
            f32x4 d = *(const f32x4*)&lds_table[b * DIM + lane * 4];
            f32x4* dst = (f32x4*)(out + (base + (long long)r) * DIM + (long long)(lane * 4));
            __builtin_nontemporal_store(d, dst);
        }
    }
}

extern "C" void kernel_launch(void* const* d_in, const int* in_sizes, int n_in,
                              void* d_out, int out_size, void* d_ws, size_t ws_size,
                              hipStream_t stream) {
    const float* time_value  = (const float*)d_in[0];
    const float* bin_edges   = (const float*)d_in[1];
    const float* embed_table = (const float*)d_in[2];
    float* out = (float*)d_out;
    int n = in_sizes[0];

    long long waves_needed  = ((long long)n + 31) / 32;
    long long blocks_needed = (waves_needed + 7) / 8;     // 8 waves / 256-thread block
    int blocks = 2048;                                    // grid-stride fills the device
    if (blocks_needed < blocks) blocks = (int)blocks_needed;
    if (blocks < 1) blocks = 1;

    duration_embed_kernel<<<blocks, 256, 0, stream>>>(time_value, bin_edges,
                                                      embed_table, out, n);
}